// BQWarp_79714593013902
// MI455X (gfx1250) — compile-verified
//
#include <hip/hip_runtime.h>

// Ball query: 32768 grid queries vs 8192 cloud points, radius 0.25, first K=10 hits
// by index order. Distance matrix computed 16x16 per wave via V_WMMA_F32_16X16X4_F32
// (K=3 padded to 4: A row = (qx,qy,qz,1), B col = (-2px,-2py,-2pz,|p|^2), epilogue
// adds |q|^2). Selection via wave32 ballots (16-bit per-row masks in index order).

#define NP   8192
#define NQ   32768
#define KSEL 10
#define R2   0.0625f   // 0.25^2

typedef float v2f __attribute__((ext_vector_type(2)));
typedef float v8f __attribute__((ext_vector_type(8)));

__global__ __launch_bounds__(256) void bq_wmma_kernel(
    const float* __restrict__ x,      // 8192 x 3 cloud points (p2)
    const float* __restrict__ q,      // 32768 x 3 grid queries (p1)
    int*   __restrict__ mapping,      // 32768 x 10
    float* __restrict__ outpts) {     // 32768 x 10 x 3
  // Cache the whole cloud (+ norms) in LDS: 8192 * float4 = 128 KB (WGP has 320 KB).
  __shared__ float4 pts[NP];
  const int tid = threadIdx.x;
  for (int i = tid; i < NP; i += 256) {
    float px = x[3 * i + 0];
    float py = x[3 * i + 1];
    float pz = x[3 * i + 2];
    pts[i] = make_float4(px, py, pz, px * px + py * py + pz * pz);
  }
  __syncthreads();

  const int lane  = tid & 31;
  const int wave  = tid >> 5;          // 8 waves/block, 16 query rows each
  const int m     = lane & 15;         // row within wave tile / col within chunk
  const int khalf = lane >> 4;         // which K-half of A/B this lane supplies
  const int rowg  = blockIdx.x * 128 + wave * 16 + m;

  // This lane's query row (lanes 0-15 and 16-31 mirror the same 16 rows).
  const float qx = q[3 * rowg + 0];
  const float qy = q[3 * rowg + 1];
  const float qz = q[3 * rowg + 2];
  const float qn = qx * qx + qy * qy + qz * qz;

  // A matrix 16x4, f32: lanes 0-15 hold K=0 (VGPR0) and K=1 (VGPR1),
  // lanes 16-31 hold K=2 and K=3.  A row = (qx, qy, qz, 1).
  v2f a;
  a.x = khalf ? qz : qx;
  a.y = khalf ? 1.0f : qy;

  // rowNorm[v]: |q|^2 of the row that D-VGPR v holds in this lane
  // (row = v + 8*(lane>>4)); loop-invariant, fetched once via lane shuffles.
  float rn[8];
#pragma unroll
  for (int v = 0; v < 8; ++v) rn[v] = __shfl(qn, v + 8 * khalf, 32);

  int cnt = (lane < 16) ? 0 : KSEL;    // lanes >=16 never own a row
  const int rowbase = rowg * KSEL;

  for (int chunk = 0; chunk < NP / 16; ++chunk) {
    // Uniform early exit once all 16 rows of this wave have K hits.
    if (__builtin_amdgcn_ballot_w32(cnt < KSEL) == 0u) break;

    // B matrix 4x16, f32: column n = (-2px, -2py, -2pz, |p|^2).
    float4 p = pts[chunk * 16 + m];
    v2f b;
    b.x = khalf ? (-2.0f * p.z) : (-2.0f * p.x);
    b.y = khalf ? p.w : (-2.0f * p.y);

    // D = A x B + 0  ->  -2<q,p> + |p|^2  per (row, col)
    v8f c = {0.f, 0.f, 0.f, 0.f, 0.f, 0.f, 0.f, 0.f};
    v8f d = __builtin_amdgcn_wmma_f32_16x16x4_f32(
        /*neg_a=*/false, a, /*neg_b=*/false, b,
        /*c_mod=*/(short)0, c, /*reuse_a=*/false, /*reuse_b=*/false);

    // One ballot per D-VGPR: bits[15:0] = row v validity (cols 0-15, in index
    // order), bits[31:16] = row v+8 validity.
    unsigned int bal[8];
#pragma unroll
    for (int v = 0; v < 8; ++v) {
      float d2 = d[v] + rn[v];
      bal[v] = __builtin_amdgcn_ballot_w32(d2 <= R2);
    }

    // Lane r owns row r: pop set bits (= ascending point index) until K found.
    if (lane < 16 && cnt < KSEL) {
      const int j = lane & 7;
      unsigned int s0 = (j & 1) ? bal[1] : bal[0];
      unsigned int s1 = (j & 1) ? bal[3] : bal[2];
      unsigned int s2 = (j & 1) ? bal[5] : bal[4];
      unsigned int s3 = (j & 1) ? bal[7] : bal[6];
      unsigned int t0 = (j & 2) ? s1 : s0;
      unsigned int t1 = (j & 2) ? s3 : s2;
      unsigned int bsel = (j & 4) ? t1 : t0;
      unsigned int m16 = (lane < 8) ? (bsel & 0xFFFFu) : (bsel >> 16);
      while (m16 && cnt < KSEL) {
        int n = __builtin_ctz(m16);
        m16 &= m16 - 1u;
        int idx = chunk * 16 + n;
        mapping[rowbase + cnt] = idx;
        float4 pp = pts[idx];
        float* o = outpts + (rowbase + cnt) * 3;
        o[0] = pp.x;
        o[1] = pp.y;
        o[2] = pp.z;
        ++cnt;
      }
    }
  }

  // Zero-fill unfound slots (harness poisons d_out with 0xAA).
  if (lane < 16) {
    for (; cnt < KSEL; ++cnt) {
      mapping[rowbase + cnt] = 0;
      float* o = outpts + (rowbase + cnt) * 3;
      o[0] = 0.f;
      o[1] = 0.f;
      o[2] = 0.f;
    }
  }
}

extern "C" void kernel_launch(void* const* d_in, const int* in_sizes, int n_in,
                              void* d_out, int out_size, void* d_ws, size_t ws_size,
                              hipStream_t stream) {
  (void)in_sizes; (void)n_in; (void)out_size; (void)d_ws; (void)ws_size;
  const float* x = (const float*)d_in[0];        // (1, 8192, 3)
  const float* q = (const float*)d_in[1];        // (1, 64, 32, 16, 3) -> 32768 x 3
  int*   mapping = (int*)d_out;                  // first 32768*10 elements (int32 bits)
  float* outpts  = (float*)d_out + NQ * KSEL;    // next 32768*10*3 floats
  bq_wmma_kernel<<<NQ / 128, 256, 0, stream>>>(x, q, mapping, outpts);
}